// SparseLinear_66314295050387
// MI455X (gfx1250) — compile-verified
//
#include <hip/hip_runtime.h>
#include <hip/hip_bf16.h>

typedef float v2f __attribute__((ext_vector_type(2)));
typedef float v8f __attribute__((ext_vector_type(8)));

#define BATCH 256
#define ROWS_PER_TILE 16
#define PAD 8u                 // bin length padding (entries); keeps 16B-aligned s_load_b128
#define COLMASK 0x3FFFu        // IN_FEATURES = 16384 -> col fits in 14 bits

// ---------------------------------------------------------------------------
// Pass 0a: zero the per-bin counters
// ---------------------------------------------------------------------------
__global__ void sl_zero_counts(unsigned* __restrict__ count, int nbins) {
    int i = blockIdx.x * blockDim.x + threadIdx.x;
    if (i < nbins) count[i] = 0u;
}

// ---------------------------------------------------------------------------
// Pass 0b: zero the (padded) value array so pad slots contribute exactly 0
// ---------------------------------------------------------------------------
__global__ void sl_zero_vals(float4* __restrict__ p, int len4) {
    int i = blockIdx.x * blockDim.x + threadIdx.x;
    if (i < len4) p[i] = make_float4(0.f, 0.f, 0.f, 0.f);
}

// ---------------------------------------------------------------------------
// Pass 1: histogram nnz per 16-row bin
// ---------------------------------------------------------------------------
__global__ void sl_hist(const long long* __restrict__ indices,
                        unsigned* __restrict__ count, int nnz) {
    int k = blockIdx.x * blockDim.x + threadIdx.x;
    if (k < nnz) {
        int row = (int)indices[k];            // indices[0][k]
        atomicAdd(&count[row >> 4], 1u);
    }
}

// ---------------------------------------------------------------------------
// Pass 2: exclusive prefix scan over PADDED bin sizes (single block, LDS)
// ---------------------------------------------------------------------------
__global__ void sl_scan(const unsigned* __restrict__ count,
                        unsigned* __restrict__ start,
                        unsigned* __restrict__ cursor, int nbins) {
    __shared__ unsigned tmp[1024];
    int t = threadIdx.x;
    unsigned padded = 0u;
    if (t < nbins) padded = (count[t] + (PAD - 1u)) & ~(PAD - 1u);
    tmp[t] = padded;
    __syncthreads();
    for (int off = 1; off < 1024; off <<= 1) {
        unsigned v = (t >= off) ? tmp[t - off] : 0u;
        __syncthreads();
        tmp[t] += v;
        __syncthreads();
    }
    if (t < nbins) {
        unsigned excl = tmp[t] - padded;      // 8-aligned start
        start[t]  = excl;
        cursor[t] = excl;
    }
}

// ---------------------------------------------------------------------------
// Pass 3: scatter nnz into binned storage: meta = (rowlocal<<16)|col, val
// ---------------------------------------------------------------------------
__global__ void sl_scatter(const long long* __restrict__ indices,
                           const float* __restrict__ values,
                           unsigned* __restrict__ cursor,
                           unsigned* __restrict__ meta,
                           float* __restrict__ sval, int nnz) {
    int k = blockIdx.x * blockDim.x + threadIdx.x;
    if (k < nnz) {
        int row = (int)indices[k];
        int col = (int)indices[nnz + k];      // indices[1][k]
        unsigned pos = atomicAdd(&cursor[row >> 4], 1u);
        meta[pos] = ((unsigned)(row & 15) << 16) | (unsigned)col;
        sval[pos] = values[k];
    }
}

// ---------------------------------------------------------------------------
// One K=4 WMMA step:
//   A (16x4 sparse selector): A[m,k] = (rowlocal_k == m) ? v_k : 0
//   B (4x16 gathered x rows): B[k,n] = x[col_k, colbase+n]
// A layout: lanes 0-15 hold {K0,K1} for M=lane; lanes 16-31 hold {K2,K3}.
// ---------------------------------------------------------------------------
__device__ __forceinline__ v8f sl_wmma_chunk(v8f acc, uint4 mt, float4 vv,
                                             bool hi, int m, int colbase,
                                             const float* __restrict__ x) {
    unsigned mA0 = hi ? mt.z : mt.x;
    unsigned mA1 = hi ? mt.w : mt.y;
    float    vA0 = hi ? vv.z : vv.x;
    float    vA1 = hi ? vv.w : vv.y;

    v2f A;
    A.x = (((mA0 >> 16) & 15u) == (unsigned)m) ? vA0 : 0.0f;
    A.y = (((mA1 >> 16) & 15u) == (unsigned)m) ? vA1 : 0.0f;

    unsigned c0 = mA0 & COLMASK;              // always in-bounds, even for pad slots
    unsigned c1 = mA1 & COLMASK;
    v2f B;
    B.x = x[c0 * BATCH + colbase + m];
    B.y = x[c1 * BATCH + colbase + m];

    return __builtin_amdgcn_wmma_f32_16x16x4_f32(
        false, A, false, B, (short)0, acc, false, false);
}

// ---------------------------------------------------------------------------
// Pass 4: SpMM. One block per 16-row tile; 16 wave32s, each owns a 16x16
// output tile in WMMA f32 accumulator registers. Branch-free inner loop:
// 8 nnz (2 WMMAs) per iteration, metadata via aligned s_load_b128 pairs.
// ---------------------------------------------------------------------------
__global__ __launch_bounds__(512) void sl_spmm(
        const float* __restrict__ x,
        const float* __restrict__ bias,
        const unsigned* __restrict__ start,
        const unsigned* __restrict__ count,
        const unsigned* __restrict__ meta,
        const float* __restrict__ sval,
        float* __restrict__ out) {
    const int bin  = blockIdx.x;
    const int lane = threadIdx.x & 31;
    const int wave = threadIdx.x >> 5;        // 0..15, each owns 16 batch cols
    const int colbase = wave * 16;

    const unsigned s    = start[bin];                          // multiple of 8
    const unsigned n    = count[bin];
    const unsigned npad = (n + (PAD - 1u)) & ~(PAD - 1u);      // multiple of 8
    const unsigned nchunk = npad >> 2;                         // multiple of 2

    const int  m  = lane & 15;
    const bool hi = lane >= 16;

    const uint4*  mp = (const uint4*)(meta + s);   // 16B aligned (s % 8 == 0)
    const float4* vp = (const float4*)(sval + s);

    v8f acc = {};
    for (unsigned c = 0; c < nchunk; c += 2) {
        uint4  m0 = mp[c],     m1 = mp[c + 1];
        float4 v0 = vp[c],     v1 = vp[c + 1];
        acc = sl_wmma_chunk(acc, m0, v0, hi, m, colbase, x);
        acc = sl_wmma_chunk(acc, m1, v1, hi, m, colbase, x);
    }

    // Store 16x16 tile: VGPR r: lanes 0-15 -> M=r, lanes 16-31 -> M=8+r
    const int rowtile = bin * ROWS_PER_TILE;
    const int ncol    = colbase + m;
#pragma unroll
    for (int r = 0; r < 8; ++r) {
        int grow = rowtile + (hi ? 8 + r : r);
        out[grow * BATCH + ncol] = acc[r] + bias[grow];
    }
}

// ---------------------------------------------------------------------------
extern "C" void kernel_launch(void* const* d_in, const int* in_sizes, int n_in,
                              void* d_out, int out_size, void* d_ws, size_t ws_size,
                              hipStream_t stream) {
    const float*     x       = (const float*)d_in[0];
    const float*     values  = (const float*)d_in[1];
    const long long* indices = (const long long*)d_in[2];
    const float*     bias    = (const float*)d_in[3];

    const int nnz          = in_sizes[1];
    const int out_features = in_sizes[3];            // bias has OUT_FEATURES elems
    const int nbins        = out_features / ROWS_PER_TILE;
    const int capacity     = nnz + (int)PAD * nbins; // padded storage bound

    // Workspace layout
    unsigned* count  = (unsigned*)d_ws;              // nbins
    unsigned* start  = count  + nbins;               // nbins
    unsigned* cursor = start  + nbins;               // nbins
    unsigned* meta   = cursor + nbins;               // capacity (pad stays stale; masked)
    float*    sval   = (float*)(meta + capacity);    // capacity (pad zeroed)

    float* out = (float*)d_out;

    const int cap4 = capacity / 4;                   // capacity is a multiple of 4

    sl_zero_counts<<<(nbins + 255) / 256, 256, 0, stream>>>(count, nbins);
    sl_zero_vals<<<(cap4 + 255) / 256, 256, 0, stream>>>((float4*)sval, cap4);
    sl_hist<<<(nnz + 255) / 256, 256, 0, stream>>>(indices, count, nnz);
    sl_scan<<<1, 1024, 0, stream>>>(count, start, cursor, nbins);
    sl_scatter<<<(nnz + 255) / 256, 256, 0, stream>>>(indices, values, cursor,
                                                      meta, sval, nnz);
    sl_spmm<<<nbins, 512, 0, stream>>>(x, bias, start, count, meta, sval, out);
}